// SoftCrossEntropy_69595650064809
// MI455X (gfx1250) — compile-verified
//
#include <hip/hip_runtime.h>
#include <cstdint>

// ---------------------------------------------------------------------------
// SoftCrossEntropy (thresholded masked-mean CE) for MI455X / gfx1250.
// inputs: [B=4096, C=50257] f32, target: [B] int.
// One-pass online softmax per row (base-2 domain), HBM-bound (~823MB / 23.3TB/s).
// Data path: async global->LDS, 4-deep ring with 3 tiles in flight per wave
// (ASYNCcnt pipeline, NT cache policy so the one-shot stream skips L2 thrash).
// Raw v_exp_f32 / v_log_f32 transcendentals (args <= 0; denorm tail ~ 0).
// Deterministic: fixed reduction trees, no float atomics.
// ---------------------------------------------------------------------------

#define BLOCK 256          // 8 wave32s per workgroup
#define WAVES (BLOCK / 32)
#define NBUF  4            // staging ring depth
#define PRE   3            // max tiles in flight per wave

__device__ __constant__ const float kL2E = 1.4426950408889634f;  // log2(e)
__device__ __constant__ const float kLN2 = 0.6931471805599453f;  // ln(2)
#define THRESH 0.8f

#if defined(__has_builtin)
# if __has_builtin(__builtin_amdgcn_global_load_async_to_lds_b128)
#  define HAVE_ASYNC_LOAD 1
# endif
# if __has_builtin(__builtin_amdgcn_s_wait_asynccnt)
#  define HAVE_WAIT_ASYNC 1
# endif
#endif

// Raw transcendentals: v_exp_f32 (2^x) and v_log_f32 (log2 x).
__device__ __forceinline__ float E2(float x) {
#if defined(__has_builtin) && __has_builtin(__builtin_amdgcn_exp2f)
  return __builtin_amdgcn_exp2f(x);
#else
  return exp2f(x);
#endif
}
__device__ __forceinline__ float LG2(float x) {
#if defined(__has_builtin) && __has_builtin(__builtin_amdgcn_logf)
  return __builtin_amdgcn_logf(x);
#else
  return __log2f(x);
#endif
}

// Builtin-matching pointer types: v4i in global (AS1) and LDS (AS3).
typedef int v4i __attribute__((ext_vector_type(4)));
typedef v4i __attribute__((address_space(1)))* v4i_gptr;
typedef v4i __attribute__((address_space(3)))* v4i_lptr;

template <int N>
__device__ __forceinline__ void wait_asynccnt() {
#if defined(HAVE_WAIT_ASYNC)
  __builtin_amdgcn_s_wait_asynccnt(N);
#else
  asm volatile("s_wait_asynccnt %0" :: "i"(N) : "memory");
#endif
}

__device__ __forceinline__ void async_load_b128(const float4* gsrc, float4* lds_dst) {
#if defined(HAVE_ASYNC_LOAD)
  // (v4i AS1* global src, v4i AS3* LDS dst, imm offset, imm cpol).
  // cpol = 1 -> TH=NT: stream-once data (823MB >> 192MB L2).
  __builtin_amdgcn_global_load_async_to_lds_b128(
      (v4i_gptr)(uintptr_t)gsrc,
      (v4i_lptr)(uint32_t)(uintptr_t)lds_dst,
      0, 1);
#else
  (void)gsrc; (void)lds_dst;
#endif
}

__global__ __launch_bounds__(BLOCK)
void ce_rows_kernel(const float* __restrict__ inputs,
                    const int*   __restrict__ target,
                    float* __restrict__ row_ce,
                    float* __restrict__ row_cnt,
                    int C) {
  const int row = blockIdx.x;
  const int tid = threadIdx.x;
  const size_t rowStart = (size_t)row * (size_t)C;
  const float* rp = inputs + rowStart;

  // Hoist the target-logit gather: two dependent loads overlap the whole stream.
  int   tgt = 0;
  float xt  = 0.0f;
  if (tid == 0) {
    tgt = target[row];
    xt  = rp[tgt];
  }

  // Align to 16B: rows are only 4B aligned (C odd).
  const int lead = (int)((4u - ((uint32_t)rowStart & 3u)) & 3u);
  const int n4 = (C - lead) >> 2;                 // # of aligned float4s
  const int tailStart = lead + (n4 << 2);
  const int tailCnt = C - tailStart;              // 0..3 trailing scalars
  const float4* rp4 = (const float4*)(rp + lead);

  // Dual online-softmax accumulators (base-2 domain) for ILP.
  float mA = -INFINITY, sA = 0.0f;
  float mB = -INFINITY, sB = 0.0f;

  auto accum1 = [&](float xv) {                   // edge scalars -> chain A
    float y = xv * kL2E;
    float nm = fmaxf(mA, y);
    sA = fmaf(sA, E2(mA - nm), E2(y - nm));
    mA = nm;
  };
  auto accum4 = [&](float4 x) {
    float y0 = x.x * kL2E, y1 = x.y * kL2E, y2 = x.z * kL2E, y3 = x.w * kL2E;
    float nA = fmaxf(mA, fmaxf(y0, y1));
    float nB = fmaxf(mB, fmaxf(y2, y3));
    sA = fmaf(sA, E2(mA - nA), E2(y0 - nA) + E2(y1 - nA));
    sB = fmaf(sB, E2(mB - nB), E2(y2 - nB) + E2(y3 - nB));
    mA = nA; mB = nB;
  };

  // Edge scalars (lead + tail), distributed to distinct threads.
  if (tid < lead) accum1(rp[tid]);
  {
    int ti = tid - 32;
    if (ti >= 0 && ti < tailCnt) accum1(rp[tailStart + ti]);
  }

  const int fullTiles = n4 / BLOCK;               // 49 for C=50257
  const int remV4 = n4 - fullTiles * BLOCK;

#if defined(HAVE_ASYNC_LOAD)
  // ---- async global->LDS stream: 4-buffer ring, up to PRE tiles in flight ----
  __shared__ float4 stage[NBUF][BLOCK];           // 4 x 4KB
  auto issue = [&](int t) {
    async_load_b128(rp4 + (size_t)t * BLOCK + tid, &stage[t & (NBUF - 1)][tid]);
  };
  int issued = 0;
  for (; issued < fullTiles && issued < PRE; ++issued) issue(issued);
  for (int t = 0; t < fullTiles; ++t) {
    if (issued < fullTiles) { issue(issued); ++issued; }
    // In-order completion per wave: tile t is done once ASYNCcnt <= issued-(t+1).
    const int allow = issued - t - 1;             // uniform scalar; PRE in steady state
    if (allow >= 3)      wait_asynccnt<3>();
    else if (allow == 2) wait_asynccnt<2>();
    else if (allow == 1) wait_asynccnt<1>();
    else                 wait_asynccnt<0>();
    asm volatile("" ::: "memory");                // order DS reads after wait
    accum4(stage[t & (NBUF - 1)][tid]);           // each lane reads its own b128
  }
#else
  // ---- fallback: direct b128 streaming with software prefetch ----
  for (int t = 0; t < fullTiles; ++t) {
    const float4* p = rp4 + (size_t)t * BLOCK + tid;
    if (t + 1 < fullTiles)
      __builtin_prefetch(rp4 + (size_t)(t + 1) * BLOCK + tid, 0, 0);
    accum4(*p);
  }
#endif
  if (tid < remV4) accum4(rp4[(size_t)fullTiles * BLOCK + tid]);

  // ---- merge dual accumulators ----
  float m = fmaxf(mA, mB);
  float s = sA * E2(mA - m) + sB * E2(mB - m);

  // ---- wave32 reduction of (m, s) ----
  #pragma unroll
  for (int off = 16; off > 0; off >>= 1) {
    float mo = __shfl_xor(m, off, 32);
    float so = __shfl_xor(s, off, 32);
    float nm = fmaxf(m, mo);
    s = fmaf(s, E2(m - nm), so * E2(mo - nm));
    m = nm;
  }

  // ---- cross-wave reduction via LDS ----
  __shared__ float wm[WAVES];
  __shared__ float ws[WAVES];
  const int wave = tid >> 5;
  if ((tid & 31) == 0) { wm[wave] = m; ws[wave] = s; }
  __syncthreads();

  if (tid == 0) {
    float M = wm[0], S = ws[0];
    #pragma unroll
    for (int w = 1; w < WAVES; ++w) {
      float nm = fmaxf(M, wm[w]);
      S = fmaf(S, E2(M - nm), ws[w] * E2(wm[w] - nm));
      M = nm;
    }
    // log2 p(target) = y_t - (M + log2 S);  ln p = log2 p * ln2
    const float logp2 = xt * kL2E - M - LG2(S);
    const float prob = E2(logp2);
    const float ce = -logp2 * kLN2;
    const bool msk = prob < THRESH;
    row_ce[row]  = msk ? ce : 0.0f;
    row_cnt[row] = msk ? 1.0f : 0.0f;
  }
}

__global__ __launch_bounds__(1024)
void ce_final_kernel(const float* __restrict__ row_ce,
                     const float* __restrict__ row_cnt,
                     float* __restrict__ out, int n) {
  __shared__ float sc[1024];
  __shared__ float sn[1024];
  const int tid = threadIdx.x;
  float c = 0.0f, k = 0.0f;
  for (int i = tid; i < n; i += 1024) { c += row_ce[i]; k += row_cnt[i]; }
  sc[tid] = c; sn[tid] = k;
  __syncthreads();
  #pragma unroll
  for (int off = 512; off > 0; off >>= 1) {
    if (tid < off) { sc[tid] += sc[tid + off]; sn[tid] += sn[tid + off]; }
    __syncthreads();
  }
  if (tid == 0) {
    const float total = sc[0], cnt = sn[0];
    out[0] = (cnt > 0.0f) ? total / fmaxf(cnt, 1.0f) : total;
  }
}

extern "C" void kernel_launch(void* const* d_in, const int* in_sizes, int n_in,
                              void* d_out, int out_size, void* d_ws, size_t ws_size,
                              hipStream_t stream) {
  const float* inputs = (const float*)d_in[0];
  const int*   target = (const int*)d_in[1];
  const int B = in_sizes[1];                 // 4096 rows
  const int C = in_sizes[0] / B;             // 50257 classes

  float* row_ce  = (float*)d_ws;             // [B]
  float* row_cnt = row_ce + B;               // [B]  (32 KB total of ws)

  ce_rows_kernel<<<B, BLOCK, 0, stream>>>(inputs, target, row_ce, row_cnt, C);
  ce_final_kernel<<<1, 1024, 0, stream>>>(row_ce, row_cnt, (float*)d_out, B);
}